// S1Net_25288767438966
// MI455X (gfx1250) — compile-verified
//
#include <hip/hip_runtime.h>
#include <math.h>
#include <stdint.h>

// Izhikevich spiking-network scan for MI455X (gfx1250, wave32).
// N=1M independent neurons, 100-step serial scan each -> VALU-latency bound.
// 4 neurons/thread (float4 coalesced) gives 4 independent FMA chains/lane.
// FR/W tiles staged via CDNA5 async global->LDS (ASYNCcnt) path.

#define TPB    256
#define KCONN  4
#define SIMT   100

#define ASG __attribute__((address_space(1)))
#define ASL __attribute__((address_space(3)))

typedef __attribute__((ext_vector_type(4))) int v4i;

#if __has_builtin(__builtin_amdgcn_global_load_async_to_lds_b128)
#define HAVE_ASYNC_LDS 1
#else
#define HAVE_ASYNC_LDS 0
#warning "no __builtin_amdgcn_global_load_async_to_lds_b128 -- falling back to direct global loads"
#endif

#if __has_builtin(__builtin_amdgcn_tanhf)
#define FAST_TANH(x) __builtin_amdgcn_tanhf(x)
#else
#define FAST_TANH(x) tanhf(x)
#warning "no __builtin_amdgcn_tanhf -- falling back to ocml tanhf"
#endif

__device__ __forceinline__ void async_g2l_b128(const void* g, void* l) {
#if HAVE_ASYNC_LDS
  __builtin_amdgcn_global_load_async_to_lds_b128(
      (ASG v4i*)(uintptr_t)g,            // global source (AS1, typed v4i*)
      (ASL v4i*)(uint32_t)(uintptr_t)l,  // LDS dest: low 32 bits of generic ptr == LDS byte addr
      /*imm offset*/ 0, /*cpol*/ 0);
#endif
}

__device__ __forceinline__ void wait_async0() {
#if __has_builtin(__builtin_amdgcn_s_wait_asynccnt)
  __builtin_amdgcn_s_wait_asynccnt(0);
#else
  asm volatile("s_wait_asynccnt 0" ::: "memory");
#endif
}

__global__ __launch_bounds__(TPB)
void izh_scan_kernel(const float* __restrict__ input,
                     const float* __restrict__ FR,
                     const float* __restrict__ Cs,
                     const float* __restrict__ W,
                     float* __restrict__ out,
                     int n) {
#if HAVE_ASYNC_LDS
  __shared__ float sFR[KCONN][TPB * 4];   // 16 KB
  __shared__ float sW [KCONN][TPB * 4];   // 16 KB
#endif
  const int t  = threadIdx.x;
  const int g4 = blockIdx.x * TPB + t;            // float4 group index
  const long long base = (long long)g4 * 4;       // first neuron of this thread
  const bool active = (base + 3) < (long long)n;  // n is a multiple of 4 (N=1e6)

  float frw[4] = {0.f, 0.f, 0.f, 0.f};
  float4 in4 = make_float4(0.f, 0.f, 0.f, 0.f);

  if (active) {
#if HAVE_ASYNC_LDS
    // Kick off async global->LDS DMA for this thread's FR/W slices (no VGPR dests).
    #pragma unroll
    for (int k = 0; k < KCONN; ++k) {
      async_g2l_b128(FR + (long long)k * n + base, &sFR[k][4 * t]);
      async_g2l_b128(W  + (long long)k * n + base, &sW [k][4 * t]);
    }
#endif
    in4 = reinterpret_cast<const float4*>(input)[g4];  // overlaps with async DMA
#if HAVE_ASYNC_LDS
    wait_async0();  // each thread reads only its own 16B -> no barrier needed
    #pragma unroll
    for (int k = 0; k < KCONN; ++k) {
      const float4 f = *reinterpret_cast<const float4*>(&sFR[k][4 * t]);
      const float4 w = *reinterpret_cast<const float4*>(&sW [k][4 * t]);
      frw[0] = fmaf(f.x, w.x, frw[0]);
      frw[1] = fmaf(f.y, w.y, frw[1]);
      frw[2] = fmaf(f.z, w.z, frw[2]);
      frw[3] = fmaf(f.w, w.w, frw[3]);
    }
#else
    #pragma unroll
    for (int k = 0; k < KCONN; ++k) {
      const float4 f = reinterpret_cast<const float4*>(FR + (long long)k * n)[g4];
      const float4 w = reinterpret_cast<const float4*>(W  + (long long)k * n)[g4];
      frw[0] = fmaf(f.x, w.x, frw[0]);
      frw[1] = fmaf(f.y, w.y, frw[1]);
      frw[2] = fmaf(f.z, w.z, frw[2]);
      frw[3] = fmaf(f.w, w.w, frw[3]);
    }
#endif
  }
  if (!active) return;

  const float c = Cs[0];  // uniform -> scalar load
  const float iv[4] = {in4.x, in4.y, in4.z, in4.w};

  float in_n[4], Ic[4], v[4], u[4], spk[4];
  #pragma unroll
  for (int j = 0; j < 4; ++j) {
    const float sf  = fmaxf(FAST_TANH(frw[j]), 0.f);                 // relu(tanh)
    const float inn = fmaxf(fmaf(c, sf - iv[j], iv[j]), 0.f);        // -C*(in-sf)+in
    in_n[j] = inn;
    Ic[j]   = rintf(inn * 20.f) + 140.f;  // I + 140, step-invariant (rndne == jnp.round)
    v[j] = -70.f;  u[j] = -14.f;  spk[j] = 0.f;
  }

  // 100-step Izhikevich scan: 4 independent dependence chains per lane.
  #pragma unroll 2
  for (int s = 0; s < SIMT; ++s) {
    #pragma unroll
    for (int j = 0; j < 4; ++j) {
      // v' = v + 0.04 v^2 + 5v + 140 - u + I  ==  v*(0.04v + 6) + (Ic - u)
      const float vv = fmaf(v[j], fmaf(0.04f, v[j], 6.0f), Ic[j] - u[j]);
      // u' = u + a*(b*v' - u), with updated v (BrainCog semantics)
      const float uu = fmaf(0.02f, fmaf(0.2f, vv, -u[j]), u[j]);
      const bool sp = vv > 30.f;
      spk[j] = sp ? spk[j] + 1.f : spk[j];
      v[j]   = sp ? -55.f        : vv;
      u[j]   = sp ? uu + 2.f     : uu;
    }
  }

  const float4 frn = make_float4(spk[0] * 0.01f, spk[1] * 0.01f,
                                 spk[2] * 0.01f, spk[3] * 0.01f);
  const float4 inn4 = make_float4(in_n[0], in_n[1], in_n[2], in_n[3]);
  reinterpret_cast<float4*>(out)[g4]     = frn;   // FR_n  (first output)
  reinterpret_cast<float4*>(out + n)[g4] = inn4;  // input_n (second output; n*4B is 16B-aligned)
}

extern "C" void kernel_launch(void* const* d_in, const int* in_sizes, int n_in,
                              void* d_out, int out_size, void* d_ws, size_t ws_size,
                              hipStream_t stream) {
  (void)n_in; (void)out_size; (void)d_ws; (void)ws_size;
  const float* input = (const float*)d_in[0];  // [N]
  const float* FR    = (const float*)d_in[1];  // [4, N]
  const float* C     = (const float*)d_in[2];  // [1]
  const float* W     = (const float*)d_in[3];  // [4, N]
  float* out = (float*)d_out;                  // [2N] = FR_n ++ input_n

  const int n      = in_sizes[0];
  const int groups = n / 4;                        // N divisible by 4 (1e6)
  const int blocks = (groups + TPB - 1) / TPB;
  izh_scan_kernel<<<blocks, TPB, 0, stream>>>(input, FR, C, W, out, n);
}